// UpdateGraph_v2_29025388986859
// MI455X (gfx1250) — compile-verified
//
#include <hip/hip_runtime.h>
#include <stdint.h>

// ---------------------------------------------------------------------------
// UpdateGraph_v2: prob_emo[i] = prod_j w(E[i,j]; col j), out = prob / ||prob||_1
// Memory-bound streaming kernel for MI455X (gfx1250, wave32).
// 128 MB mandatory read  ->  ~5.5 us at 23.3 TB/s HBM.  No GEMM content, so
// no WMMA; we exercise the CDNA5 async global->LDS path + global_prefetch.
// ---------------------------------------------------------------------------

typedef float vf4 __attribute__((ext_vector_type(4)));
typedef int   v4i __attribute__((ext_vector_type(4)));

#define BLOCK   256
#define NB1     1024          // persistent blocks for the main kernel
#define EPS     1e-5f
#define THRESH  0.6f

#if defined(__has_builtin)
#  if __has_builtin(__builtin_amdgcn_global_load_async_to_lds_b128)
#    define HAVE_ASYNC_LDS 1
#  endif
#endif

__global__ __launch_bounds__(BLOCK) void emo_prod_kernel(
    const float* __restrict__ prob_all_au,   // [64]
    const float* __restrict__ cpt,           // [rows, 64]
    const float* __restrict__ prob_AU_full,  // [64] (cancels under L1 norm)
    float* __restrict__ out,                 // [rows]  un-normalized products
    float* __restrict__ partial,             // [gridDim.x] block |.| sums
    int rows)
{
  __shared__ __align__(16) float s_raw[128]; // [0..63]=p, [64..127]=pA
  __shared__ float s_occ[64];                // 1.0f if occluded else 0.0f
  __shared__ float s_red[BLOCK];

  const int tid = threadIdx.x;

  // ---- Stage the 512B parameter tables to LDS via the CDNA5 async path ----
#ifdef HAVE_ASYNC_LDS
  if (tid < 32) {
    const float* g = (tid < 16) ? (prob_all_au + tid * 4)
                                : (prob_AU_full + (tid - 16) * 4);
    float* l = (tid < 16) ? (s_raw + tid * 4) : (s_raw + 64 + (tid - 16) * 4);
    __builtin_amdgcn_global_load_async_to_lds_b128(
        (__attribute__((address_space(1))) v4i*)(v4i*)g,
        (__attribute__((address_space(3))) v4i*)(v4i*)l,
        /*offset=*/0, /*cpol=*/0);
  }
#  if __has_builtin(__builtin_amdgcn_s_wait_asynccnt)
  __builtin_amdgcn_s_wait_asynccnt(0);
#  else
  asm volatile("s_wait_asynccnt 0x0" ::: "memory");
#  endif
#else
  if (tid < 64) {
    s_raw[tid]      = prob_all_au[tid];
    s_raw[64 + tid] = prob_AU_full[tid];
  }
#endif
  __syncthreads();

  if (tid < 64) {
    // occ_j = (p_j > 0.6).  au_j factors cancel in the final normalization.
    s_occ[tid] = (s_raw[tid] > THRESH) ? 1.0f : 0.0f;
  }
  __syncthreads();

  const int total = (int)gridDim.x * BLOCK;
  float absacc = 0.0f;

  for (int r = (int)blockIdx.x * BLOCK + tid; r < rows; r += total) {
    const vf4* rp = (const vf4*)(cpt + (size_t)r * 64);

    // Prefetch the next row this thread will process (gfx1250 global_prefetch).
    if (r + total < rows)
      __builtin_prefetch(cpt + (size_t)(r + total) * 64, 0, 1);

    float pr[4] = {1.0f, 1.0f, 1.0f, 1.0f};

#pragma unroll
    for (int c = 0; c < 16; ++c) {
      vf4 v = __builtin_nontemporal_load(rp + c);   // read-once 128 MB stream
#pragma unroll
      for (int i = 0; i < 4; ++i) {
        const int j = c * 4 + i;
        float x = v[i];
        float e = (x > 0.0f) ? x : EPS;             // E = where(cpt>0,cpt,eps)
        float ne = 1.0f - e;
        if (j < 32) ne = (ne > 0.0f) ? ne : EPS;    // negE1 = where(1-E>0,...)
        float w = (s_occ[j] != 0.0f) ? e : ne;      // occ selects branch
        pr[i] *= w;
      }
    }
    float prob = (pr[0] * pr[1]) * (pr[2] * pr[3]);
    out[r] = prob;
    absacc += fabsf(prob);
  }

  // Deterministic block tree reduction of sum(|prob|)
  s_red[tid] = absacc;
  __syncthreads();
  for (int w = BLOCK / 2; w > 0; w >>= 1) {
    if (tid < w) s_red[tid] += s_red[tid + w];
    __syncthreads();
  }
  if (tid == 0) partial[blockIdx.x] = s_red[0];
}

__global__ __launch_bounds__(BLOCK) void denom_kernel(
    const float* __restrict__ partial, int n, float* __restrict__ denom)
{
  __shared__ float s[BLOCK];
  float a = 0.0f;
  for (int i = threadIdx.x; i < n; i += BLOCK) a += partial[i];
  s[threadIdx.x] = a;
  __syncthreads();
  for (int w = BLOCK / 2; w > 0; w >>= 1) {
    if (threadIdx.x < w) s[threadIdx.x] += s[threadIdx.x + w];
    __syncthreads();
  }
  if (threadIdx.x == 0) denom[0] = fmaxf(s[0], 1e-12f);
}

__global__ __launch_bounds__(BLOCK) void scale_kernel(
    float* __restrict__ out, const float* __restrict__ denom, int n)
{
  int i = (int)blockIdx.x * BLOCK + threadIdx.x;
  float d = denom[0];
  if (i < n) out[i] = out[i] / d;
}

extern "C" void kernel_launch(void* const* d_in, const int* in_sizes, int n_in,
                              void* d_out, int out_size, void* d_ws, size_t ws_size,
                              hipStream_t stream)
{
  const float* prob_all_au  = (const float*)d_in[0];
  const float* cpt          = (const float*)d_in[1];
  const float* prob_AU_full = (const float*)d_in[2];
  float* out = (float*)d_out;
  float* ws  = (float*)d_ws;      // ws[0]=denom, ws[1..nb] = block partials

  const int rows = in_sizes[1] / 64;
  if (rows <= 0) return;

  int nb = NB1;
  int maxb = (rows + BLOCK - 1) / BLOCK;
  if (nb > maxb) nb = maxb;
  int cap = (int)(ws_size / sizeof(float)) - 1;
  if (cap > 0 && nb > cap) nb = cap;
  if (nb < 1) nb = 1;

  float* partial = ws + 1;

  emo_prod_kernel<<<nb, BLOCK, 0, stream>>>(prob_all_au, cpt, prob_AU_full,
                                            out, partial, rows);
  denom_kernel<<<1, BLOCK, 0, stream>>>(partial, nb, ws);
  scale_kernel<<<(rows + BLOCK - 1) / BLOCK, BLOCK, 0, stream>>>(out, ws, rows);
}